// PostProcessor_4672924418523
// MI455X (gfx1250) — compile-verified
//
#include <hip/hip_runtime.h>

#define WW   320
#define HH   96
#define HW   (WW * HH)     // 30720
#define NB   64
#define NC   3
#define KDET 50
#define PI_F 3.14159265358979323846f

typedef __attribute__((ext_vector_type(2))) float v2f;
typedef __attribute__((ext_vector_type(8))) float v8f;

__device__ __forceinline__ unsigned long long umax64(unsigned long long a, unsigned long long b) {
    return a > b ? a : b;
}

// Analytic 3x3 inverse (adjugate / det)
__device__ __forceinline__ void inv3(const float* __restrict__ m, float* o) {
    float a=m[0],b=m[1],c=m[2],d=m[3],e=m[4],f=m[5],g=m[6],h=m[7],i=m[8];
    float A  =  e*i - f*h;
    float Bc = -(d*i - f*g);
    float Cc =  d*h - e*g;
    float det = a*A + b*Bc + c*Cc;
    float inv = 1.0f / det;
    o[0] =  A * inv;          o[1] = -(b*i - c*h) * inv;  o[2] =  (b*f - c*e) * inv;
    o[3] =  Bc * inv;         o[4] =  (a*i - c*g) * inv;  o[5] = -(a*f - c*d) * inv;
    o[6] =  Cc * inv;         o[7] = -(a*h - b*g) * inv;  o[8] =  (a*e - b*d) * inv;
}

// ---------------------------------------------------------------------------
// Kernel 1: 3x3 NMS + per-(b,c) top-50. One block per (b,c) slice.
// 1024 threads * 30 elements each = 30720 = H*W.
// Key packing: value bits in high 32, (0xFFFFFFFF - idx) in low 32 so that
// u64-argmax breaks score ties toward the LOWER spatial index (top_k order).
// ---------------------------------------------------------------------------
__global__ __launch_bounds__(1024) void k_nms_topk(const float* __restrict__ heat,
                                                   unsigned long long* __restrict__ keys) {
    const int bc  = blockIdx.x;                       // 0..191
    const float* hp = heat + (size_t)bc * HW;
    const int tid = threadIdx.x;
    constexpr int PT = HW / 1024;                     // 30
    unsigned long long kk[PT];

    #pragma unroll
    for (int e = 0; e < PT; ++e) {
        int idx = e * 1024 + tid;                     // coalesced
        int y = idx / WW;
        int x = idx - y * WW;
        float c = hp[idx];
        float m = c;
        bool xl = (x > 0), xr = (x < WW - 1);
        if (xl)          m = fmaxf(m, hp[idx - 1]);
        if (xr)          m = fmaxf(m, hp[idx + 1]);
        if (y > 0) {
                         m = fmaxf(m, hp[idx - WW]);
            if (xl)      m = fmaxf(m, hp[idx - WW - 1]);
            if (xr)      m = fmaxf(m, hp[idx - WW + 1]);
        }
        if (y < HH - 1) {
                         m = fmaxf(m, hp[idx + WW]);
            if (xl)      m = fmaxf(m, hp[idx + WW - 1]);
            if (xr)      m = fmaxf(m, hp[idx + WW + 1]);
        }
        float v = (c == m) ? c : 0.0f;                // NMS: keep only local maxima
        kk[e] = ((unsigned long long)__float_as_uint(v) << 32)
              | (unsigned long long)(0xFFFFFFFFu - (unsigned)idx);
    }

    __shared__ unsigned long long red[32];
    __shared__ unsigned long long bmax;
    const int lane = tid & 31;
    const int wave = tid >> 5;

    for (int it = 0; it < KDET; ++it) {
        unsigned long long m = 0ull;
        #pragma unroll
        for (int e = 0; e < PT; ++e) m = umax64(m, kk[e]);
        #pragma unroll
        for (int off = 16; off > 0; off >>= 1)
            m = umax64(m, __shfl_xor(m, off, 32));
        if (lane == 0) red[wave] = m;
        __syncthreads();
        if (wave == 0) {
            unsigned long long t = red[lane];
            #pragma unroll
            for (int off = 16; off > 0; off >>= 1)
                t = umax64(t, __shfl_xor(t, off, 32));
            if (lane == 0) bmax = t;
        }
        __syncthreads();
        unsigned long long M = bmax;
        if (tid == 0) keys[(size_t)bc * KDET + it] = M;
        // Keys are unique (index embedded); zero the selected one without
        // dynamic register indexing.
        #pragma unroll
        for (int e = 0; e < PT; ++e) if (kk[e] == M) kk[e] = 0ull;
    }
}

// ---------------------------------------------------------------------------
// Kernel 2: per-batch merge top-50 of 150 + regression gather + geometry.
// One block (128 threads, 4 waves) per batch. The shared linear transform
// locs = depth * (K^-1 @ T^-1) @ proj_h runs on V_WMMA_F32_16X16X4_F32,
// 16 detections per wave.
// ---------------------------------------------------------------------------
__global__ __launch_bounds__(128) void k_detect(const float* __restrict__ reg,
                                                const unsigned long long* __restrict__ keys,
                                                const float* __restrict__ trans_mat,
                                                const float* __restrict__ K_mat,
                                                const float* __restrict__ img_size,
                                                float* __restrict__ out) {
    const int b    = blockIdx.x;
    const int tid  = threadIdx.x;
    const int lane = tid & 31;
    const int wave = tid >> 5;

    __shared__ unsigned long long cand[NC * KDET];    // 150 merge candidates
    __shared__ float sScore[KDET];
    __shared__ int   sCls[KDET];
    __shared__ int   sInd[KDET];
    __shared__ float sM3[9];                          // K^-1 @ T^-1
    __shared__ float sK[9];
    __shared__ float sImg[2];
    __shared__ unsigned long long red[4];
    __shared__ unsigned long long bmax;

    // Load candidates; repack low bits so ties break by ascending i2 = c*K + k
    for (int i = tid; i < NC * KDET; i += 128) {
        unsigned long long k0 = keys[((size_t)b * NC + (i / KDET)) * KDET + (i % KDET)];
        cand[i] = (k0 & 0xFFFFFFFF00000000ull)
                | (unsigned long long)(NC * KDET - 1 - i);
    }
    if (tid == 0) {
        float ti[9], ki[9];
        inv3(trans_mat, ti);
        inv3(K_mat, ki);
        #pragma unroll
        for (int r = 0; r < 3; ++r)
            #pragma unroll
            for (int c = 0; c < 3; ++c)
                sM3[r*3+c] = ki[r*3+0]*ti[0*3+c] + ki[r*3+1]*ti[1*3+c] + ki[r*3+2]*ti[2*3+c];
        #pragma unroll
        for (int j = 0; j < 9; ++j) sK[j] = K_mat[j];
        sImg[0] = img_size[0];
        sImg[1] = img_size[1];
    }
    __syncthreads();

    // ---- merge: top-50 of 150 ----
    for (int it = 0; it < KDET; ++it) {
        unsigned long long m = cand[tid];
        if (tid + 128 < NC * KDET) m = umax64(m, cand[tid + 128]);
        #pragma unroll
        for (int off = 16; off > 0; off >>= 1)
            m = umax64(m, __shfl_xor(m, off, 32));
        if (lane == 0) red[wave] = m;
        __syncthreads();
        if (tid == 0) {
            unsigned long long t = umax64(umax64(red[0], red[1]), umax64(red[2], red[3]));
            bmax = t;
            int i2  = (NC * KDET - 1) - (int)(t & 0xFFFFFFFFull);
            int c   = i2 / KDET;
            int kk  = i2 - c * KDET;
            unsigned long long orig = keys[((size_t)b * NC + c) * KDET + kk];
            sScore[it] = __uint_as_float((unsigned)(t >> 32));
            sCls[it]   = c;
            sInd[it]   = (int)(0xFFFFFFFFu - (unsigned)(orig & 0xFFFFFFFFull));
        }
        __syncthreads();
        unsigned long long M = bmax;
        if (cand[tid] == M) cand[tid] = 0ull;
        if (tid + 128 < NC * KDET && cand[tid + 128] == M) cand[tid + 128] = 0ull;
        __syncthreads();
    }

    // ---- geometry: wave w owns detections w*16 .. w*16+15 ----
    const int col = lane & 15;
    const int n   = wave * 16 + col;                  // 0..63, mask n<50 at stores
    const int nn  = (n < KDET) ? n : (KDET - 1);      // clamp for safe shared reads
    const int ind = sInd[nn];
    const int cls = sCls[nn];

    const float* rb = reg + (size_t)b * 8 * HW + (size_t)ind;  // channel stride = HW
    float p0 = rb[0*HW], p1 = rb[1*HW], p2 = rb[2*HW], p3 = rb[3*HW];
    float p4 = rb[4*HW], p5 = rb[5*HW], p6 = rb[6*HW], p7 = rb[7*HW];

    float xs = (float)(ind % WW);
    float ys = (float)(ind / WW);
    float projx = xs + p1;
    float projy = ys + p2;

    // WMMA operands (f32 16x16x4): A = M3 in rows 0..2 (zero-padded),
    // B cols = proj_h per detection. Per-ISA layout: VGPR0 carries K={0,2}
    // (lanes 0-15 / 16-31), VGPR1 carries K={1,3}.
    v2f Av, Bv;
    {
        int  m  = lane & 15;
        bool lo = lane < 16;
        Av.x = (m < 3) ? sM3[m * 3 + (lo ? 0 : 2)] : 0.0f;   // K=0 or K=2 column
        Av.y = (lo && m < 3) ? sM3[m * 3 + 1] : 0.0f;        // K=1 (K=3 pad = 0)
        Bv.x = lo ? projx : 1.0f;                            // K=0 row / K=2 row (=1)
        Bv.y = lo ? projy : 0.0f;                            // K=1 row / K=3 row (=0)
    }
    v8f D = {};
    // D = A x B + 0  (EXEC is full here; masking happens only at the stores)
    D = __builtin_amdgcn_wmma_f32_16x16x4_f32(false, Av, false, Bv, (short)0, D,
                                              false, false);
    // Lanes 0-15: D[0..2] = (M3 @ proj_h) x,y,z for detection n.

    if (lane < 16 && n < KDET) {
        float depth = p0 * 16.32f + 28.01f;
        float lx = depth * D[0];
        float ly = depth * D[1];
        float lz = depth * D[2];

        const float dimref[3][3] = {{3.88f, 1.63f, 1.53f},
                                    {1.78f, 1.70f, 0.58f},
                                    {0.88f, 1.73f, 0.67f}};
        float dl = __expf(p3) * dimref[cls][0];
        float dh = __expf(p4) * dimref[cls][1];
        float dw = __expf(p5) * dimref[cls][2];
        ly += 0.5f * dh;

        float rays   = atanf(lx / (lz + 1e-7f));
        float alphas = atanf(p6 / (p7 + 1e-7f)) + (p7 >= 0.0f ? -PI_F * 0.5f : PI_F * 0.5f);
        float roty   = alphas + rays;
        roty = (roty >  PI_F) ? roty - 2.0f * PI_F : roty;
        roty = (roty < -PI_F) ? roty + 2.0f * PI_F : roty;

        float cs = __cosf(roty), sn = __sinf(roty);
        const float xc[8] = {-0.5f, 0.5f, 0.5f, 0.5f, 0.5f, -0.5f, -0.5f, -0.5f};
        const float yc[8] = {-1.f, -1.f, 0.f, 0.f, -1.f, -1.f, 0.f, 0.f};
        const float zc[8] = {-0.5f, -0.5f, -0.5f, 0.5f, 0.5f, 0.5f, 0.5f, -0.5f};
        float umin = 1e30f, umax = -1e30f, vmin = 1e30f, vmax = -1e30f;
        #pragma unroll
        for (int j = 0; j < 8; ++j) {
            float cxx = dl * xc[j], cyy = dh * yc[j], czz = dw * zc[j];
            float bx =  cs * cxx + sn * czz + lx;     // ry @ corner + loc
            float by =  cyy + ly;
            float bz = -sn * cxx + cs * czz + lz;
            float ix = sK[0]*bx + sK[1]*by + sK[2]*bz;
            float iy = sK[3]*bx + sK[4]*by + sK[5]*bz;
            float iz = sK[6]*bx + sK[7]*by + sK[8]*bz;
            float u = ix / iz, v = iy / iz;
            umin = fminf(umin, u); umax = fmaxf(umax, u);
            vmin = fminf(vmin, v); vmax = fmaxf(vmax, v);
        }
        float x0 = fminf(fmaxf(umin, 0.0f), sImg[0]);
        float x1 = fminf(fmaxf(umax, 0.0f), sImg[0]);
        float y0 = fminf(fmaxf(vmin, 0.0f), sImg[1]);
        float y1 = fminf(fmaxf(vmax, 0.0f), sImg[1]);

        // Output: [boxes (B,K,7)] [scores (B,K,1)] [labels (B,K,1)] [box2d (B,K,4)]
        size_t base = (size_t)b * KDET + n;
        float* pb = out + base * 7;
        pb[0] = lx; pb[1] = ly; pb[2] = lz;
        pb[3] = dh; pb[4] = dw; pb[5] = dl;          // dims rolled: [h, w, l]
        pb[6] = roty;
        out[(size_t)NB * KDET * 7 + base] = sScore[nn];
        out[(size_t)NB * KDET * 8 + base] = (float)cls;
        float* b2 = out + (size_t)NB * KDET * 9 + base * 4;
        b2[0] = x0; b2[1] = y0; b2[2] = x1; b2[3] = y1;
    }
}

extern "C" void kernel_launch(void* const* d_in, const int* in_sizes, int n_in,
                              void* d_out, int out_size, void* d_ws, size_t ws_size,
                              hipStream_t stream) {
    (void)in_sizes; (void)n_in; (void)out_size; (void)ws_size;
    const float* heat = (const float*)d_in[0];   // (64,3,96,320)
    const float* reg  = (const float*)d_in[1];   // (64,8,96,320)
    const float* tm   = (const float*)d_in[2];   // 3x3
    const float* km   = (const float*)d_in[3];   // 3x3
    const float* isz  = (const float*)d_in[4];   // 2
    float* out = (float*)d_out;
    unsigned long long* keys = (unsigned long long*)d_ws;  // 192*50 packed keys

    k_nms_topk<<<NB * NC, 1024, 0, stream>>>(heat, keys);
    k_detect<<<NB, 128, 0, stream>>>(reg, keys, tm, km, isz, out);
}